// MyGAT_82935818485899
// MI455X (gfx1250) — compile-verified
//
#include <hip/hip_runtime.h>
#include <math.h>

typedef __attribute__((ext_vector_type(2))) float v2f;
typedef __attribute__((ext_vector_type(8))) float v8f;

#define NEG_SLOPE 0.2f
#define NN 4096

// ---------------------------------------------------------------------------
// GEMM: C[M,Nc] = A[M,K] @ B[K,Nc] + bias, fp32 WMMA 16x16x4.
// One wave per 16x16 C tile; 4 waves (128 thr) per block stacked along M.
// Layouts per CDNA5 ISA 7.12.2:
//   A 16x4 f32 : lanes 0-15 -> M=lane,  VGPR{0,1} = K{0,1}; lanes 16-31 K{2,3}
//   B  4x16 f32: lanes 0-15 -> N=lane,  VGPR{0,1} = K{0,1}; lanes 16-31 K{2,3}
//   C/D        : VGPR i -> M = i + 8*(lane>>4), N = lane&15
// ---------------------------------------------------------------------------
__global__ __launch_bounds__(128)
void gat_gemm_bias_wmma(const float* __restrict__ A, const float* __restrict__ B,
                        const float* __restrict__ bias, float* __restrict__ C,
                        int M, int K, int Nc) {
  const int lane = threadIdx.x & 31;
  const int wave = threadIdx.x >> 5;
  const int tn = blockIdx.x << 4;                 // col tile base
  const int tm = ((blockIdx.y << 2) + wave) << 4; // row tile base
  const int r  = lane & 15;
  const int kk = (lane >> 4) << 1;                // 0 or 2

  const float* Arow = A + (size_t)(tm + r) * K + kk;
  const float* Bcol = B + (size_t)kk * Nc + (tn + r);

  v8f acc = {};
  for (int k0 = 0; k0 < K; k0 += 4) {
    v2f a, b;
    a.x = Arow[k0];
    a.y = Arow[k0 + 1];
    b.x = Bcol[(size_t)k0 * Nc];
    b.y = Bcol[(size_t)(k0 + 1) * Nc];
    acc = __builtin_amdgcn_wmma_f32_16x16x4_f32(false, a, false, b, (short)0,
                                                acc, false, false);
  }
  const float bv = bias[tn + r];
  const int rbase = tm + ((lane >> 4) << 3);
#pragma unroll
  for (int i = 0; i < 8; ++i)
    C[(size_t)(rbase + i) * Nc + tn + r] = acc[i] + bv;
}

// ---------------------------------------------------------------------------
// Per-node attention scores: s1[i] = h[i,:].a[:D], s2[i] = h[i,:].a[D:]
// One wave per node.
// ---------------------------------------------------------------------------
__global__ __launch_bounds__(32)
void gat_node_scores(const float* __restrict__ h, const float* __restrict__ avec,
                     float* __restrict__ s1, float* __restrict__ s2, int D) {
  const int i = blockIdx.x;
  const int lane = threadIdx.x;
  const float* hr = h + (size_t)i * D;
  float a1 = 0.f, a2 = 0.f;
  for (int d = lane; d < D; d += 32) {
    float hv = hr[d];
    a1 += hv * avec[d];
    a2 += hv * avec[D + d];
  }
#pragma unroll
  for (int off = 16; off; off >>= 1) {
    a1 += __shfl_xor(a1, off, 32);
    a2 += __shfl_xor(a2, off, 32);
  }
  if (lane == 0) { s1[i] = a1; s2[i] = a2; }
}

// ---------------------------------------------------------------------------
// CSR construction (rebuilt each call; edge_index is constant input)
// ---------------------------------------------------------------------------
__global__ void gat_zero_ints(int* __restrict__ p, int n) {
  int i = blockIdx.x * blockDim.x + threadIdx.x;
  if (i < n) p[i] = 0;
}
__global__ void gat_hist_src(const int* __restrict__ src, int E, int* __restrict__ cnt) {
  int e = blockIdx.x * blockDim.x + threadIdx.x;
  if (e < E) atomicAdd(&cnt[src[e]], 1);
}
__global__ void gat_scan_rows(const int* __restrict__ cnt, int* __restrict__ rp,
                              int* __restrict__ cur, int n) {
  if (threadIdx.x == 0 && blockIdx.x == 0) {
    int s = 0;
    rp[0] = 0;
    for (int i = 0; i < n; ++i) { cur[i] = s; s += cnt[i]; rp[i + 1] = s; }
  }
}
__global__ void gat_scatter_cols(const int* __restrict__ src, const int* __restrict__ dst,
                                 int E, int* __restrict__ cur, int* __restrict__ col) {
  int e = blockIdx.x * blockDim.x + threadIdx.x;
  if (e < E) {
    int p = atomicAdd(&cur[src[e]], 1);
    col[p] = dst[e];
  }
}

// ---------------------------------------------------------------------------
// Row softmax over sparse logits + aggregation + ELU.
// One 256-thread block per row i. Dense LDS row collapses duplicate edges
// exactly like .at[src,dst].set (duplicates carry identical values).
// out[i,:] = ELU( softmax_j( lrelu(s1[i]+s2[j]+ab) ) @ h )
// ---------------------------------------------------------------------------
__global__ __launch_bounds__(256)
void gat_row_attn_elu(const float* __restrict__ h, const float* __restrict__ s1,
                      const float* __restrict__ s2, const float* __restrict__ abp,
                      int abi, const int* __restrict__ rp, const int* __restrict__ col,
                      float* __restrict__ out, int D) {
  __shared__ float p[NN];
  __shared__ float red[256];
  const int i = blockIdx.x;
  const int tid = threadIdx.x;
  const float NEG_INF = -__builtin_inff();

  for (int j = tid; j < NN; j += 256) p[j] = NEG_INF;
  __syncthreads();

  const float ab  = abp[abi];
  const float s1i = s1[i];
  const int beg = rp[i], end = rp[i + 1];
  for (int k = beg + tid; k < end; k += 256) {
    int j = col[k];
    p[j] = s1i + s2[j] + ab;   // duplicates write identical values
  }
  __syncthreads();

  // leaky relu (keeps -inf) + row max
  float m = NEG_INF;
  for (int j = tid; j < NN; j += 256) {
    float x = p[j];
    x = x > 0.f ? x : NEG_SLOPE * x;
    p[j] = x;
    m = fmaxf(m, x);
  }
  red[tid] = m;
  __syncthreads();
  for (int s = 128; s; s >>= 1) {
    if (tid < s) red[tid] = fmaxf(red[tid], red[tid + s]);
    __syncthreads();
  }
  m = red[0];
  __syncthreads();

  // exp + row sum (unnormalized weights left in p[])
  float sum = 0.f;
  for (int j = tid; j < NN; j += 256) {
    float x = p[j];
    float t = (x == NEG_INF) ? 0.f : __expf(x - m);
    p[j] = t;
    sum += t;
  }
  red[tid] = sum;
  __syncthreads();
  for (int s = 128; s; s >>= 1) {
    if (tid < s) red[tid] += red[tid + s];
    __syncthreads();
  }
  const float inv = 1.0f / red[0];
  __syncthreads();

  // out[i,d] = ELU( inv * sum_j p[j] * h[j,d] ); skip zero weights (uniform branch)
  for (int d = tid; d < D; d += 256) {
    float acc = 0.f;
    for (int j = 0; j < NN; ++j) {
      float w = p[j];
      if (w != 0.f) acc += w * h[(size_t)j * D + d];
    }
    acc *= inv;
    out[(size_t)i * D + d] = acc > 0.f ? acc : (__expf(acc) - 1.0f);
  }
}

// ---------------------------------------------------------------------------
// Elementwise helpers
// ---------------------------------------------------------------------------
__global__ void gat_mean4(const float* __restrict__ a, const float* __restrict__ b,
                          const float* __restrict__ c, const float* __restrict__ d,
                          float* __restrict__ o, int n) {
  int i = blockIdx.x * blockDim.x + threadIdx.x;
  if (i < n) o[i] = 0.25f * (a[i] + b[i] + c[i] + d[i]);
}
__global__ void gat_lrelu(float* __restrict__ x, int n) {
  int i = blockIdx.x * blockDim.x + threadIdx.x;
  if (i < n) {
    float v = x[i];
    x[i] = v > 0.f ? v : NEG_SLOPE * v;
  }
}

// ---------------------------------------------------------------------------
extern "C" void kernel_launch(void* const* d_in, const int* in_sizes, int n_in,
                              void* d_out, int out_size, void* d_ws, size_t ws_size,
                              hipStream_t stream) {
  (void)n_in; (void)out_size; (void)ws_size;
  const int N = NN, IN = 512, HID = 256, OUT = 128, H = 4;
  const int E = in_sizes[1] / 2;

  const float* x      = (const float*)d_in[0];
  const int*   ei     = (const int*)  d_in[1];
  const float* l0_hW  = (const float*)d_in[2];
  const float* l0_hWb = (const float*)d_in[3];
  const float* l0_ha  = (const float*)d_in[4];
  const float* l0_hab = (const float*)d_in[5];
  const float* l0_oW  = (const float*)d_in[6];
  const float* l0_oWb = (const float*)d_in[7];
  const float* l0_oa  = (const float*)d_in[8];
  const float* l0_oab = (const float*)d_in[9];
  const float* l1_hW  = (const float*)d_in[10];
  const float* l1_hWb = (const float*)d_in[11];
  const float* l1_ha  = (const float*)d_in[12];
  const float* l1_hab = (const float*)d_in[13];
  const float* l1_oW  = (const float*)d_in[14];
  const float* l1_oWb = (const float*)d_in[15];
  const float* l1_oa  = (const float*)d_in[16];
  const float* l1_oab = (const float*)d_in[17];
  const float* lin_W  = (const float*)d_in[18];
  const float* lin_b  = (const float*)d_in[19];

  const int* src = ei;
  const int* dst = ei + E;

  // workspace carve-up (256B aligned)
  char* wp = (char*)d_ws;
  auto carve = [&](size_t bytes) -> void* {
    void* p = (void*)wp;
    wp += (bytes + 255) & ~(size_t)255;
    return p;
  };
  int* cnt = (int*)carve((size_t)N * 4);
  int* rp  = (int*)carve((size_t)(N + 1) * 4);
  int* cur = (int*)carve((size_t)N * 4);
  int* col = (int*)carve((size_t)E * 4);
  const size_t FB = (size_t)N * HID * sizeof(float);
  float* B0 = (float*)carve(FB);  // h buffer
  float* B1 = (float*)carve(FB);  // head y0 / mean
  float* B2 = (float*)carve(FB);  // head y1 / act1
  float* B3 = (float*)carve(FB);  // head y2
  float* B4 = (float*)carve(FB);  // head y3
  float* B5 = (float*)carve(FB);  // act0
  float* s1 = (float*)carve((size_t)N * 4);
  float* s2 = (float*)carve((size_t)N * 4);
  float* Y[4] = {B1, B2, B3, B4};

  // ---- CSR build ----
  gat_zero_ints<<<(N + 255) / 256, 256, 0, stream>>>(cnt, N);
  gat_hist_src<<<(E + 255) / 256, 256, 0, stream>>>(src, E, cnt);
  gat_scan_rows<<<1, 1, 0, stream>>>(cnt, rp, cur, N);
  gat_scatter_cols<<<(E + 255) / 256, 256, 0, stream>>>(src, dst, E, cur, col);

  // one GAT sub-layer: h = xin@W + Wb ; scores ; sparse softmax + aggr + ELU
  auto gat_layer = [&](const float* xin, int K, const float* W, const float* Wb,
                       const float* a, const float* abp, int abi, float* ybuf) {
    dim3 g(HID / 16, N / 64);
    gat_gemm_bias_wmma<<<g, 128, 0, stream>>>(xin, W, Wb, B0, N, K, HID);
    gat_node_scores<<<N, 32, 0, stream>>>(B0, a, s1, s2, HID);
    gat_row_attn_elu<<<N, 256, 0, stream>>>(B0, s1, s2, abp, abi, rp, col, ybuf, HID);
  };
  const int NE = N * HID;

  // ---- multihead layer 0 (in 512 -> 256) ----
  for (int t = 0; t < H; ++t)
    gat_layer(x, IN, l0_hW + (size_t)t * IN * HID, l0_hWb + (size_t)t * HID,
              l0_ha + (size_t)t * 2 * HID, l0_hab, t, Y[t]);
  gat_mean4<<<(NE + 255) / 256, 256, 0, stream>>>(B1, B2, B3, B4, B1, NE);
  gat_layer(B1, HID, l0_oW, l0_oWb, l0_oa, l0_oab, 0, B5);
  gat_lrelu<<<(NE + 255) / 256, 256, 0, stream>>>(B5, NE);

  // ---- multihead layer 1 (256 -> 256) ----
  for (int t = 0; t < H; ++t)
    gat_layer(B5, HID, l1_hW + (size_t)t * HID * HID, l1_hWb + (size_t)t * HID,
              l1_ha + (size_t)t * 2 * HID, l1_hab, t, Y[t]);
  gat_mean4<<<(NE + 255) / 256, 256, 0, stream>>>(B1, B2, B3, B4, B1, NE);
  gat_layer(B1, HID, l1_oW, l1_oWb, l1_oa, l1_oab, 0, B2);
  gat_lrelu<<<(NE + 255) / 256, 256, 0, stream>>>(B2, NE);

  // ---- final linear 256 -> 128 ----
  dim3 gf(OUT / 16, N / 64);
  gat_gemm_bias_wmma<<<gf, 128, 0, stream>>>(B2, lin_W, lin_b, (float*)d_out,
                                             N, HID, OUT);
}